// OhemBCELoss_75969381532179
// MI455X (gfx1250) — compile-verified
//
#include <hip/hip_runtime.h>
#include <stdint.h>

#define N_ELEM   16777216u      // 16*1024*1024
#define N_MIN_K  1048576u
#define THRESH_F 0.35667494393873245f
#define NBLK     1024
#define BT       256

// ws layout (uint32 words):
//  [0..1023]    hist[4][256]
//  [1024]       PREFIX (radix-select prefix / final nth bits)
//  [1025]       RANK   (remaining descending rank)
//  [1026]       CNT_OVER
//  [1027]       CNT_GT
//  [2048..3071] partials_over (float, one per block)
//  [3072..4095] partials_gt   (float, one per block)
//  [4096..]     optional loss cache (N_ELEM floats) if ws_size allows
#define W_HIST0   0u
#define W_PREFIX  1024u
#define W_RANK    1025u
#define W_CNTOVER 1026u
#define W_CNTGT   1027u
#define W_POVER   2048u
#define W_PGT     3072u
#define W_CACHE   4096u

typedef float v2f __attribute__((ext_vector_type(2)));
typedef float v4f __attribute__((ext_vector_type(4)));
typedef float v8f __attribute__((ext_vector_type(8)));

// Wave32 sum reduction via V_WMMA_F32_16X16X4_F32:
// A(16x4): lane L<16 holds (M=L,K=0..1), lane L>=16 holds (M=L-16,K=2..3).
// With a={x,0} and B=ones, D[m][n] = x_m + x_{m+16}. D VGPR r holds row r
// (lanes 0-15) and row r+8 (lanes 16-31); sum the 8 VGPRs in-lane, then one
// xor-16 shuffle gives the full 32-lane sum in every lane.
__device__ __forceinline__ float wave_sum_wmma(float x) {
    v2f a; a.x = x;   a.y = 0.0f;
    v2f b; b.x = 1.0f; b.y = 1.0f;
    v8f c = {};
    c = __builtin_amdgcn_wmma_f32_16x16x4_f32(false, a, false, b,
                                              (short)0, c, false, false);
    float s = c[0] + c[1] + c[2] + c[3] + c[4] + c[5] + c[6] + c[7];
    s += __shfl_xor(s, 16, 32);
    return s;
}

__device__ __forceinline__ float bce_loss(float o, float t) {
    float lo  = fmaxf(logf(o),    -100.0f);
    float l1o = fmaxf(log1pf(-o), -100.0f);
    return -(t * lo + (1.0f - t) * l1o);
}

__global__ void init_ws(uint32_t* __restrict__ ws) {
    for (uint32_t i = threadIdx.x; i < 4096u; i += BT) ws[i] = 0u;
    __syncthreads();
    if (threadIdx.x == 0) ws[W_RANK] = N_MIN_K;   // 0-indexed descending rank
}

// Pass 0: loss + level-0 histogram + (loss > THRESH) sum/count, optional cache fill.
// When caching, inputs are read exactly once -> non-temporal loads keep the
// 67MB loss cache resident in the 192MB L2 for passes 1..4.
__global__ void pass0_hist_over(const v4f* __restrict__ out4,
                                const v4f* __restrict__ tgt4,
                                uint32_t* __restrict__ ws,
                                v4f* __restrict__ cache4,
                                int use_cache) {
    __shared__ uint32_t lh[256];
    __shared__ float    wsum[BT / 32];
    __shared__ uint32_t ccnt;
    for (uint32_t i = threadIdx.x; i < 256u; i += BT) lh[i] = 0u;
    if (threadIdx.x == 0) ccnt = 0u;
    __syncthreads();

    float    sum_over = 0.0f;
    uint32_t cnt_over = 0u;
    const uint32_t n4     = N_ELEM / 4u;
    const uint32_t stride = gridDim.x * blockDim.x;
    for (uint32_t i = blockIdx.x * blockDim.x + threadIdx.x; i < n4; i += stride) {
        v4f o, t;
        if (use_cache) {
            // streamed once, never re-read: bypass L2 retention
            o = __builtin_nontemporal_load(&out4[i]);
            t = __builtin_nontemporal_load(&tgt4[i]);
        } else {
            o = out4[i];
            t = tgt4[i];
        }
        float l0 = bce_loss(o.x, t.x);
        float l1 = bce_loss(o.y, t.y);
        float l2 = bce_loss(o.z, t.z);
        float l3 = bce_loss(o.w, t.w);
        if (use_cache) {
            v4f lv; lv.x = l0; lv.y = l1; lv.z = l2; lv.w = l3;
            cache4[i] = lv;                       // RT hint: keep in L2
        }
        atomicAdd(&lh[__float_as_uint(l0) >> 24], 1u);
        atomicAdd(&lh[__float_as_uint(l1) >> 24], 1u);
        atomicAdd(&lh[__float_as_uint(l2) >> 24], 1u);
        atomicAdd(&lh[__float_as_uint(l3) >> 24], 1u);
        if (l0 > THRESH_F) { sum_over += l0; cnt_over++; }
        if (l1 > THRESH_F) { sum_over += l1; cnt_over++; }
        if (l2 > THRESH_F) { sum_over += l2; cnt_over++; }
        if (l3 > THRESH_F) { sum_over += l3; cnt_over++; }
    }

    float wtot = wave_sum_wmma(sum_over);           // all lanes active here
    const int lane = threadIdx.x & 31;
    const int wid  = threadIdx.x >> 5;
    if (lane == 0) wsum[wid] = wtot;
    atomicAdd(&ccnt, cnt_over);
    __syncthreads();

    for (uint32_t i = threadIdx.x; i < 256u; i += BT)
        atomicAdd(&ws[W_HIST0 + i], lh[i]);
    if (threadIdx.x == 0) {
        float s = 0.0f;
        for (int w = 0; w < BT / 32; ++w) s += wsum[w];
        ((float*)ws)[W_POVER + blockIdx.x] = s;       // fixed slot: deterministic
        atomicAdd(&ws[W_CNTOVER], ccnt);
    }
}

// Levels 1..3: histogram of next byte among elements matching the prefix.
__global__ void hist_level(const v4f* __restrict__ out4,
                           const v4f* __restrict__ tgt4,
                           const v4f* __restrict__ cache4,
                           int use_cache,
                           uint32_t* __restrict__ ws,
                           int level) {
    __shared__ uint32_t lh[256];
    for (uint32_t i = threadIdx.x; i < 256u; i += BT) lh[i] = 0u;
    __syncthreads();

    const uint32_t prefix    = ws[W_PREFIX];
    const int      hi_shift  = 32 - 8 * level;   // 24,16,8
    const int      bin_shift = 24 - 8 * level;   // 16,8,0
    const uint32_t n4     = N_ELEM / 4u;
    const uint32_t stride = gridDim.x * blockDim.x;
    for (uint32_t i = blockIdx.x * blockDim.x + threadIdx.x; i < n4; i += stride) {
        float l0, l1, l2, l3;
        if (use_cache) {
            v4f lv = cache4[i];
            l0 = lv.x; l1 = lv.y; l2 = lv.z; l3 = lv.w;
        } else {
            v4f o = out4[i];
            v4f t = tgt4[i];
            l0 = bce_loss(o.x, t.x); l1 = bce_loss(o.y, t.y);
            l2 = bce_loss(o.z, t.z); l3 = bce_loss(o.w, t.w);
        }
        uint32_t b0 = __float_as_uint(l0), b1 = __float_as_uint(l1);
        uint32_t b2 = __float_as_uint(l2), b3 = __float_as_uint(l3);
        if ((b0 >> hi_shift) == prefix) atomicAdd(&lh[(b0 >> bin_shift) & 0xFFu], 1u);
        if ((b1 >> hi_shift) == prefix) atomicAdd(&lh[(b1 >> bin_shift) & 0xFFu], 1u);
        if ((b2 >> hi_shift) == prefix) atomicAdd(&lh[(b2 >> bin_shift) & 0xFFu], 1u);
        if ((b3 >> hi_shift) == prefix) atomicAdd(&lh[(b3 >> bin_shift) & 0xFFu], 1u);
    }
    __syncthreads();
    for (uint32_t i = threadIdx.x; i < 256u; i += BT)
        atomicAdd(&ws[W_HIST0 + 256u * (uint32_t)level + i], lh[i]);
}

// Scan one histogram level descending; refine prefix and remaining rank.
__global__ void select_bin(uint32_t* __restrict__ ws, int level) {
    if (threadIdx.x != 0 || blockIdx.x != 0) return;
    const uint32_t* h = ws + W_HIST0 + 256u * (uint32_t)level;
    uint32_t K = ws[W_RANK];
    uint32_t cum = 0u, b = 0u;
    for (int i = 255; i >= 0; --i) {
        uint32_t c = h[i];
        if (cum + c > K) { b = (uint32_t)i; break; }
        cum += c;
    }
    ws[W_RANK]   = K - cum;
    ws[W_PREFIX] = (ws[W_PREFIX] << 8) | b;
}

// Final streaming pass: sum/count of losses strictly greater than nth.
__global__ void gt_stats(const v4f* __restrict__ out4,
                         const v4f* __restrict__ tgt4,
                         const v4f* __restrict__ cache4,
                         int use_cache,
                         uint32_t* __restrict__ ws) {
    __shared__ float    wsum[BT / 32];
    __shared__ uint32_t ccnt;
    if (threadIdx.x == 0) ccnt = 0u;
    __syncthreads();

    const float nth = __uint_as_float(ws[W_PREFIX]);
    float    sum_gt = 0.0f;
    uint32_t cnt_gt = 0u;
    const uint32_t n4     = N_ELEM / 4u;
    const uint32_t stride = gridDim.x * blockDim.x;
    for (uint32_t i = blockIdx.x * blockDim.x + threadIdx.x; i < n4; i += stride) {
        float l0, l1, l2, l3;
        if (use_cache) {
            v4f lv = cache4[i];
            l0 = lv.x; l1 = lv.y; l2 = lv.z; l3 = lv.w;
        } else {
            v4f o = out4[i];
            v4f t = tgt4[i];
            l0 = bce_loss(o.x, t.x); l1 = bce_loss(o.y, t.y);
            l2 = bce_loss(o.z, t.z); l3 = bce_loss(o.w, t.w);
        }
        if (l0 > nth) { sum_gt += l0; cnt_gt++; }
        if (l1 > nth) { sum_gt += l1; cnt_gt++; }
        if (l2 > nth) { sum_gt += l2; cnt_gt++; }
        if (l3 > nth) { sum_gt += l3; cnt_gt++; }
    }

    float wtot = wave_sum_wmma(sum_gt);
    const int lane = threadIdx.x & 31;
    const int wid  = threadIdx.x >> 5;
    if (lane == 0) wsum[wid] = wtot;
    atomicAdd(&ccnt, cnt_gt);
    __syncthreads();
    if (threadIdx.x == 0) {
        float s = 0.0f;
        for (int w = 0; w < BT / 32; ++w) s += wsum[w];
        ((float*)ws)[W_PGT + blockIdx.x] = s;
        atomicAdd(&ws[W_CNTGT], ccnt);
    }
}

__global__ void finalize(const uint32_t* __restrict__ ws, float* __restrict__ out) {
    __shared__ float sh[BT];
    __shared__ float s_over_sh;
    const float* wf = (const float*)ws;

    float s = 0.0f;
    for (uint32_t i = threadIdx.x; i < NBLK; i += BT) s += wf[W_POVER + i];
    sh[threadIdx.x] = s;
    __syncthreads();
    for (int off = BT / 2; off > 0; off >>= 1) {
        if (threadIdx.x < (uint32_t)off) sh[threadIdx.x] += sh[threadIdx.x + off];
        __syncthreads();
    }
    if (threadIdx.x == 0) s_over_sh = sh[0];
    __syncthreads();

    s = 0.0f;
    for (uint32_t i = threadIdx.x; i < NBLK; i += BT) s += wf[W_PGT + i];
    sh[threadIdx.x] = s;
    __syncthreads();
    for (int off = BT / 2; off > 0; off >>= 1) {
        if (threadIdx.x < (uint32_t)off) sh[threadIdx.x] += sh[threadIdx.x + off];
        __syncthreads();
    }

    if (threadIdx.x == 0) {
        float    sum_over = s_over_sh;
        float    sum_gt   = sh[0];
        float    nth      = __uint_as_float(ws[W_PREFIX]);
        uint32_t cnt_over = ws[W_CNTOVER];
        uint32_t cnt_gt   = ws[W_CNTGT];
        float result;
        if (nth > THRESH_F) {
            result = sum_over / fmaxf((float)cnt_over, 1.0f);
        } else {
            // top N_MIN = everything > nth, plus (N_MIN - cnt_gt) ties equal to nth
            result = (sum_gt + (float)(N_MIN_K - cnt_gt) * nth) / (float)N_MIN_K;
        }
        out[0] = result;
    }
}

extern "C" void kernel_launch(void* const* d_in, const int* in_sizes, int n_in,
                              void* d_out, int out_size, void* d_ws, size_t ws_size,
                              hipStream_t stream) {
    (void)in_sizes; (void)n_in; (void)out_size;
    const v4f* o4 = (const v4f*)d_in[0];
    const v4f* t4 = (const v4f*)d_in[1];
    uint32_t* ws  = (uint32_t*)d_ws;
    float*    out = (float*)d_out;

    const size_t need_cache = ((size_t)W_CACHE + (size_t)N_ELEM) * sizeof(uint32_t);
    const int use_cache = (ws_size >= need_cache) ? 1 : 0;
    v4f* cache4 = (v4f*)(ws + W_CACHE);

    init_ws<<<1, BT, 0, stream>>>(ws);
    pass0_hist_over<<<NBLK, BT, 0, stream>>>(o4, t4, ws, cache4, use_cache);
    for (int level = 1; level <= 3; ++level) {
        select_bin<<<1, 1, 0, stream>>>(ws, level - 1);
        hist_level<<<NBLK, BT, 0, stream>>>(o4, t4, cache4, use_cache, ws, level);
    }
    select_bin<<<1, 1, 0, stream>>>(ws, 3);
    gt_stats<<<NBLK, BT, 0, stream>>>(o4, t4, cache4, use_cache, ws);
    finalize<<<1, BT, 0, stream>>>(ws, out);
}